// TaskSpanGenerator2_24790551232644
// MI455X (gfx1250) — compile-verified
//
#include <hip/hip_runtime.h>
#include <cstdint>
#include <cstddef>

typedef __attribute__((ext_vector_type(2))) float v2f;
typedef __attribute__((ext_vector_type(8))) float v8f;

#define B_      16
#define L_      4096
#define D_      1024
#define W_      15
#define G_      64
#define LW_     (L_ * W_)        // 61440 spans per batch
#define NSORT_  65536            // padded pow2 per batch
#define NEGINF_ (-10000.0f)

#define CHUNK_    16384          // keys per LDS sort chunk (128 KB of 320 KB WGP LDS)
#define CTHREADS_ 1024
#define CPAIRS_   (CHUNK_ / 2 / CTHREADS_)   // 8 compare-pairs per thread per step

#define SCORES_BYTES ((size_t)B_ * L_ * 3 * sizeof(float))                 // 786,432
#define KEYS_BYTES   ((size_t)B_ * NSORT_ * sizeof(unsigned long long))    // 8,388,608

// ---------------------------------------------------------------------------
// Kernel 1: scores[b,l,c] = inputs[b,l,:] . W_bie[c,:] + b_bie[c]  via WMMA f32
// One wave = one 16-token tile. A = 16x4 f32 tile of inputs, B = 4x16 with
// cols 0..2 = W_bie rows (rest zero). Two accumulators to relax the
// WMMA->WMMA accumulation chain; software prefetch 256B ahead on the stream.
// ---------------------------------------------------------------------------
__global__ __launch_bounds__(256) void scores_wmma_kernel(
    const float* __restrict__ inp, const float* __restrict__ Wb,
    const float* __restrict__ bb, float* __restrict__ scores) {
  const int wave = (blockIdx.x * blockDim.x + threadIdx.x) >> 5;  // wave32
  const int lane = threadIdx.x & 31;
  const int m    = lane & 15;   // A: row M / B: col N / C: col N
  const int half = lane >> 4;   // K-half select
  const int tile = wave << 4;   // first token of this 16-token tile

  const float* rowA = inp + (size_t)(tile + m) * D_ + 2 * half;
  // B operand: lane n<3 feeds W_bie row n; others clamped+zeroed (no divergence)
  const float* rowB = Wb + (size_t)((m < 3) ? m : 0) * D_ + 2 * half;
  const float  bsel = (m < 3) ? 1.0f : 0.0f;

  v8f acc0 = {}, acc1 = {};
#pragma unroll 1
  for (int kk = 0; kk < D_ / 16; ++kk) {        // 64 groups of 4 K-steps
    __builtin_prefetch(rowA + 16 * kk + 64, 0, 0);  // 256B ahead on A stream
#pragma unroll
    for (int t = 0; t < 4; ++t) {
      v2f a  = *(const v2f*)(rowA + 16 * kk + 4 * t);
      v2f bv = *(const v2f*)(rowB + 16 * kk + 4 * t);
      bv.x *= bsel;
      bv.y *= bsel;
      if (t & 1)
        acc1 = __builtin_amdgcn_wmma_f32_16x16x4_f32(false, a, false, bv,
                                                     (short)0, acc1, false, false);
      else
        acc0 = __builtin_amdgcn_wmma_f32_16x16x4_f32(false, a, false, bv,
                                                     (short)0, acc0, false, false);
    }
  }
  acc0 += acc1;
  // D layout: lane n (0-15) vgpr r -> [M=r][N=n]; lane n+16 -> [M=r+8][N=n]
  if (m < 3) {
    const float bias_c = bb[m];
#pragma unroll
    for (int r = 0; r < 8; ++r) {
      const int tok = tile + 8 * half + r;
      scores[(size_t)tok * 3 + m] = acc0[r] + bias_c;
    }
  }
}

// ---------------------------------------------------------------------------
// Kernel 2: logits / mask / masked_logits / sort keys / BCE block partials.
// Block = 256 tokens of one batch; LDS-staged scores with 14-token halo.
// ---------------------------------------------------------------------------
__global__ __launch_bounds__(256) void logits_kernel(
    const float* __restrict__ scores, const float* __restrict__ bias,
    const int* __restrict__ seqlens, float* __restrict__ out,
    unsigned long long* __restrict__ keys, float* __restrict__ partials) {
  __shared__ float S[(256 + W_ - 1) * 3];
  __shared__ float sbias[W_];
  __shared__ float red[256];

  const int b   = blockIdx.x >> 4;          // 16 blocks per batch
  const int l0  = (blockIdx.x & 15) * 256;
  const int tid = threadIdx.x;

  for (int idx = tid; idx < (256 + W_ - 1) * 3; idx += 256) {
    const int tok = l0 + idx / 3;   // right-pad with zeros past L (per batch)
    S[idx] = (tok < L_) ? scores[((size_t)b * L_ + tok) * 3 + (idx % 3)] : 0.0f;
  }
  if (tid < W_) sbias[tid] = bias[tid];
  __syncthreads();

  const int seqlen = seqlens[b];
  const int l      = l0 + tid;
  const float s0   = S[tid * 3 + 0];
  float run = 0.0f, acc = 0.0f;

  float* mout = out + 1 + ((size_t)b * L_ + l) * W_;
  unsigned long long* kout = keys + (size_t)b * NSORT_ + (size_t)l * W_;

#pragma unroll
  for (int w = 0; w < W_; ++w) {
    const float logit = s0 + run + S[(tid + w) * 3 + 2] + sbias[w];
    const bool  mask  = (l + w) < seqlen;
    const float z     = mask ? logit : 0.0f;
    acc += fmaxf(z, 0.0f) + log1pf(expf(-fabsf(z)));   // t-term handled later
    const float masked = mask ? logit : (logit + NEGINF_);
    mout[w] = masked;
    // stable descending order == ascending on (sortable(-masked), index)
    unsigned int u = __float_as_uint(-masked);
    u = (u & 0x80000000u) ? ~u : (u | 0x80000000u);
    kout[w] = ((unsigned long long)u << 32) | (unsigned int)(l * W_ + w);
    if (w >= 1) run += S[(tid + w) * 3 + 1];
  }

  red[tid] = acc;
  __syncthreads();
  for (int s = 128; s > 0; s >>= 1) {
    if (tid < s) red[tid] += red[tid + s];
    __syncthreads();
  }
  if (tid == 0) partials[blockIdx.x] = red[0];
}

// Pad sort slots 61440..65535 per batch with +inf keys.
__global__ __launch_bounds__(256) void pad_keys_kernel(unsigned long long* __restrict__ keys) {
  const int t = blockIdx.x * blockDim.x + threadIdx.x;  // B_*(NSORT_-LW_)
  const int b = t / (NSORT_ - LW_);
  const int i = t % (NSORT_ - LW_);
  keys[(size_t)b * NSORT_ + LW_ + i] = ~0ull;
}

// ---------------------------------------------------------------------------
// Bitonic sort (ascending u64 keys): 16K-key LDS chunks (128 KB) minimize
// global passes: 1 local-full + 3 global + 2 local-tail kernels total.
// ---------------------------------------------------------------------------
__global__ __launch_bounds__(CTHREADS_) void bitonic_local_full(
    unsigned long long* __restrict__ keys) {
  __shared__ unsigned long long sh[CHUNK_];
  const size_t base = (size_t)blockIdx.x * CHUNK_;
  const int tid = threadIdx.x;
#pragma unroll
  for (int p = 0; p < CHUNK_ / CTHREADS_; ++p)
    sh[tid + p * CTHREADS_] = keys[base + tid + p * CTHREADS_];
  __syncthreads();
  for (int k = 2; k <= CHUNK_; k <<= 1) {
    for (int j = k >> 1; j >= 1; j >>= 1) {
#pragma unroll
      for (int p = 0; p < CPAIRS_; ++p) {
        const int t  = tid + p * CTHREADS_;          // pair index
        const int li = ((t & ~(j - 1)) << 1) | (t & (j - 1));
        const int pj = li | j;
        const int gi = (int)((base + li) & (NSORT_ - 1));
        const bool up = (gi & k) == 0;
        const unsigned long long a = sh[li], c = sh[pj];
        if ((a > c) == up) { sh[li] = c; sh[pj] = a; }
      }
      __syncthreads();
    }
  }
#pragma unroll
  for (int p = 0; p < CHUNK_ / CTHREADS_; ++p)
    keys[base + tid + p * CTHREADS_] = sh[tid + p * CTHREADS_];
}

__global__ __launch_bounds__(CTHREADS_) void bitonic_local_tail(
    unsigned long long* __restrict__ keys, int k) {
  __shared__ unsigned long long sh[CHUNK_];
  const size_t base = (size_t)blockIdx.x * CHUNK_;
  const int tid = threadIdx.x;
#pragma unroll
  for (int p = 0; p < CHUNK_ / CTHREADS_; ++p)
    sh[tid + p * CTHREADS_] = keys[base + tid + p * CTHREADS_];
  __syncthreads();
  for (int j = CHUNK_ >> 1; j >= 1; j >>= 1) {
#pragma unroll
    for (int p = 0; p < CPAIRS_; ++p) {
      const int t  = tid + p * CTHREADS_;
      const int li = ((t & ~(j - 1)) << 1) | (t & (j - 1));
      const int pj = li | j;
      const int gi = (int)((base + li) & (NSORT_ - 1));
      const bool up = (gi & k) == 0;
      const unsigned long long a = sh[li], c = sh[pj];
      if ((a > c) == up) { sh[li] = c; sh[pj] = a; }
    }
    __syncthreads();
  }
#pragma unroll
  for (int p = 0; p < CHUNK_ / CTHREADS_; ++p)
    keys[base + tid + p * CTHREADS_] = sh[tid + p * CTHREADS_];
}

__global__ __launch_bounds__(256) void bitonic_global(
    unsigned long long* __restrict__ keys, int k, int j) {
  const int t = blockIdx.x * blockDim.x + threadIdx.x;  // B_*NSORT_
  const int b = t >> 16;
  const int i = t & (NSORT_ - 1);
  const int ixj = i ^ j;
  if (ixj > i) {
    unsigned long long* base = keys + ((size_t)b << 16);
    const unsigned long long a = base[i], c = base[ixj];
    const bool up = (i & k) == 0;
    if ((a > c) == up) { base[i] = c; base[ixj] = a; }
  }
}

// ---------------------------------------------------------------------------
// Emit begins/widths (as floats, per harness output dtype) from sorted keys.
// ---------------------------------------------------------------------------
__global__ __launch_bounds__(256) void write_order_kernel(
    const unsigned long long* __restrict__ keys, float* __restrict__ out) {
  const int t = blockIdx.x * blockDim.x + threadIdx.x;  // B_*LW_
  const int b = t / LW_;
  const int r = t % LW_;
  const unsigned int idx = (unsigned int)keys[(size_t)b * NSORT_ + r];
  float* begins = out + 1 + (size_t)B_ * LW_;
  float* widths = begins + (size_t)B_ * LW_;
  begins[t] = (float)(idx / W_);
  widths[t] = (float)(idx % W_);
}

// ---------------------------------------------------------------------------
// Deterministic loss finalize: sum 256 partials, subtract gold z-terms
// (deduped to match .at[].set() semantics; masked golds contribute 0).
// ---------------------------------------------------------------------------
__global__ __launch_bounds__(256) void finalize_loss_kernel(
    const float* __restrict__ partials, const int* __restrict__ gb,
    const int* __restrict__ gw, const int* __restrict__ seqlens,
    float* __restrict__ out) {
  __shared__ float red[256];
  const int tid = threadIdx.x;
  float acc = partials[tid];
  const float* ml = out + 1;
  for (int e = tid; e < B_ * G_; e += 256) {
    const int b = e / G_, g = e % G_;
    const int begin = gb[b * G_ + g], width = gw[b * G_ + g];
    bool dup = false;
    for (int g2 = 0; g2 < g; ++g2)
      if (gb[b * G_ + g2] == begin && gw[b * G_ + g2] == width) { dup = true; break; }
    if (!dup && (begin + width) < seqlens[b])
      acc -= ml[((size_t)b * L_ + begin) * W_ + width];  // mask==1 -> masked==z
  }
  red[tid] = acc;
  __syncthreads();
  for (int s = 128; s > 0; s >>= 1) {
    if (tid < s) red[tid] += red[tid + s];
    __syncthreads();
  }
  if (tid == 0) out[0] = red[0] * 1.0f;  // WEIGHT
}

// ---------------------------------------------------------------------------
extern "C" void kernel_launch(void* const* d_in, const int* in_sizes, int n_in,
                              void* d_out, int out_size, void* d_ws, size_t ws_size,
                              hipStream_t stream) {
  const float* inp  = (const float*)d_in[0];   // [B,L,D]
  const float* Wb   = (const float*)d_in[1];   // [3,D]
  const float* bb   = (const float*)d_in[2];   // [3]
  const float* bias = (const float*)d_in[3];   // [W]
  const int*   seql = (const int*)d_in[4];     // [B]
  const int*   gb   = (const int*)d_in[5];     // [B,G]
  const int*   gw   = (const int*)d_in[6];     // [B,G]
  float* out = (float*)d_out;

  char* ws = (char*)d_ws;
  float* scores = (float*)ws;
  unsigned long long* keys = (unsigned long long*)(ws + SCORES_BYTES);
  float* partials = (float*)(ws + SCORES_BYTES + KEYS_BYTES);

  // 1) WMMA GEMM: 65536 tokens / 16 per wave = 4096 waves; 8 waves/block
  scores_wmma_kernel<<<512, 256, 0, stream>>>(inp, Wb, bb, scores);

  // 2) logits + mask + keys + BCE partials (256 blocks == partials count)
  logits_kernel<<<B_ * (L_ / 256), 256, 0, stream>>>(scores, bias, seql, out, keys, partials);

  // 3) pad sort slots
  pad_keys_kernel<<<(B_ * (NSORT_ - LW_)) / 256, 256, 0, stream>>>(keys);

  // 4) bitonic sort per batch (65536 keys): 16K-key LDS chunks, 6 launches
  const int chunks = (B_ * NSORT_) / CHUNK_;  // 64
  bitonic_local_full<<<chunks, CTHREADS_, 0, stream>>>(keys);
  for (int k = CHUNK_ * 2; k <= NSORT_; k <<= 1) {
    for (int j = k >> 1; j >= CHUNK_; j >>= 1)
      bitonic_global<<<(B_ * NSORT_) / 256, 256, 0, stream>>>(keys, k, j);
    bitonic_local_tail<<<chunks, CTHREADS_, 0, stream>>>(keys, k);
  }

  // 5) begins/widths
  write_order_kernel<<<(B_ * LW_) / 256, 256, 0, stream>>>(keys, out);

  // 6) deterministic loss
  finalize_loss_kernel<<<1, 256, 0, stream>>>(partials, gb, gw, seql, out);
}